// Erosion2d_86517821212128
// MI455X (gfx1250) — compile-verified
//
#include <hip/hip_runtime.h>
#include <stdint.h>

// Erosion (3x3 min-filter, radius 1, pad 1e9) over (16,64,256,256) fp32.
// Memory-bound: 512 MiB total traffic -> ~23 us floor at 23.3 TB/s HBM.
// Strategy: stage halo tiles into LDS with CDNA5 async global->LDS DMA
// (ASYNCcnt / s_wait_asynccnt), compute 9-point min from LDS, stream the
// result out with non-temporal 128-bit stores.

#define HW         256
#define TILE_W     64
#define TILE_H     16
#define PATCH_H    (TILE_H + 2)   // 18 halo rows
#define LDS_STRIDE 72             // floats per patch row; interior at col 4 => 16B aligned
#define PAD_VAL    1.0e9f

typedef float v4f __attribute__((ext_vector_type(4)));
typedef int   vi4 __attribute__((vector_size(16)));   // matches builtin's b128 pointee

typedef __attribute__((address_space(1))) int as1_int;
typedef __attribute__((address_space(3))) int as3_int;
typedef __attribute__((address_space(1))) vi4 as1_vi4;
typedef __attribute__((address_space(3))) vi4 as3_vi4;

__device__ __forceinline__ void async_b32(const float* g, float* l) {
#if __has_builtin(__builtin_amdgcn_global_load_async_to_lds_b32)
  __builtin_amdgcn_global_load_async_to_lds_b32(
      (as1_int*)(uintptr_t)g,
      (as3_int*)(uint32_t)(uintptr_t)l, 0, 0);
#else
  uint32_t la = (uint32_t)(uintptr_t)l;
  uint64_t ga = (uint64_t)(uintptr_t)g;
  asm volatile("global_load_async_to_lds_b32 %0, %1, off" :: "v"(la), "v"(ga) : "memory");
#endif
}

__device__ __forceinline__ void async_b128(const float* g, float* l) {
#if __has_builtin(__builtin_amdgcn_global_load_async_to_lds_b128)
  __builtin_amdgcn_global_load_async_to_lds_b128(
      (as1_vi4*)(uintptr_t)g,
      (as3_vi4*)(uint32_t)(uintptr_t)l, 0, 0);
#else
  uint32_t la = (uint32_t)(uintptr_t)l;
  uint64_t ga = (uint64_t)(uintptr_t)g;
  asm volatile("global_load_async_to_lds_b128 %0, %1, off" :: "v"(la), "v"(ga) : "memory");
#endif
}

__device__ __forceinline__ void wait_async_zero() {
#if __has_builtin(__builtin_amdgcn_s_wait_asynccnt)
  __builtin_amdgcn_s_wait_asynccnt(0);
#else
  asm volatile("s_wait_asynccnt 0" ::: "memory");
#endif
}

__global__ __launch_bounds__(256) void
erosion3x3_kernel(const float* __restrict__ in, float* __restrict__ out) {
  __shared__ float smem[PATCH_H * LDS_STRIDE];

  const int tid    = threadIdx.x;
  const int tileX0 = blockIdx.x * TILE_W;
  const int tileY0 = blockIdx.y * TILE_H;
  const long  base = (long)blockIdx.z * (HW * HW);
  const float* gplane = in + base;

  // ---- Stage interior: patch cols 1..64 (lds cols 4..67), 16 x b128 per row ----
  for (int i = tid; i < PATCH_H * (TILE_W / 4); i += 256) {
    const int row = i >> 4;        // 0..17
    const int q   = i & 15;        // 0..15 (float4 within row)
    const int gy  = tileY0 + row - 1;
    float* ldst   = &smem[row * LDS_STRIDE + 4 + q * 4];   // 16B aligned
    if ((unsigned)gy < HW) {
      async_b128(gplane + gy * HW + tileX0 + q * 4, ldst);
    } else {
      ldst[0] = PAD_VAL; ldst[1] = PAD_VAL; ldst[2] = PAD_VAL; ldst[3] = PAD_VAL;
    }
  }

  // ---- Stage halo columns: patch px=0 -> lds col 3, px=65 -> lds col 68 ----
  if (tid < 2 * PATCH_H) {
    const int row  = tid >> 1;
    const int side = tid & 1;
    const int gy   = tileY0 + row - 1;
    const int gx   = side ? (tileX0 + TILE_W) : (tileX0 - 1);
    float* ldst    = &smem[row * LDS_STRIDE + (side ? 68 : 3)];
    if ((unsigned)gy < HW && (unsigned)gx < HW) {
      async_b32(gplane + gy * HW + gx, ldst);
    } else {
      *ldst = PAD_VAL;
    }
  }

  wait_async_zero();     // drain this wave's ASYNCcnt before the barrier
  __syncthreads();

  // ---- Compute: each thread -> 4 consecutive outputs of one row ----
  // Output (oy, ox) needs lds cols ox+3..ox+5 of patch rows oy..oy+2.
  const int oy  = tid >> 4;          // 0..15
  const int ox0 = (tid & 15) * 4;    // 0,4,...,60
  float acc0, acc1, acc2, acc3;
#pragma unroll
  for (int r = 0; r < 3; ++r) {
    const float* wr = &smem[(oy + r) * LDS_STRIDE + ox0 + 3];
    const float w0 = wr[0], w1 = wr[1], w2 = wr[2];
    const float w3 = wr[3], w4 = wr[4], w5 = wr[5];
    const float m0 = fminf(fminf(w0, w1), w2);
    const float m1 = fminf(fminf(w1, w2), w3);
    const float m2 = fminf(fminf(w2, w3), w4);
    const float m3 = fminf(fminf(w3, w4), w5);
    if (r == 0) { acc0 = m0; acc1 = m1; acc2 = m2; acc3 = m3; }
    else {
      acc0 = fminf(acc0, m0); acc1 = fminf(acc1, m1);
      acc2 = fminf(acc2, m2); acc3 = fminf(acc3, m3);
    }
  }

  v4f v = {acc0, acc1, acc2, acc3};
  float* optr = out + base + (long)(tileY0 + oy) * HW + tileX0 + ox0;  // 16B aligned
  __builtin_nontemporal_store(v, (v4f*)optr);   // output never re-read: keep out of L2
}

extern "C" void kernel_launch(void* const* d_in, const int* in_sizes, int n_in,
                              void* d_out, int out_size, void* d_ws, size_t ws_size,
                              hipStream_t stream) {
  const float* x = (const float*)d_in[0];
  float* out     = (float*)d_out;
  const int planes = in_sizes[0] / (HW * HW);   // 16*64 = 1024
  dim3 grid(HW / TILE_W, HW / TILE_H, planes);  // (4, 16, 1024)
  dim3 block(256);                              // 8 wave32 waves
  hipLaunchKernelGGL(erosion3x3_kernel, grid, block, 0, stream, x, out);
}